// ConditioningEncoder_19172734009776
// MI455X (gfx1250) — compile-verified
//
#include <hip/hip_runtime.h>
#include <hip/hip_bf16.h>

// ---------------------------------------------------------------------------
// ConditioningEncoder for MI455X (gfx1250, wave32)
//   out[p, 0:128)   = LN(x_t[p])
//   out[p, 128:256) = LN(prior_mel[p])
//   out[p, 256:320) = LN(silu(f0*w1+b1) @ w2 + b2)      <-- WMMA f32 16x16x4
//   out[p, 320:384) = LN(blurred phoneme embed)
//   out[p, 384:416) = LN(voicing embed)
// ---------------------------------------------------------------------------

#define Bq   16
#define Tq   4096
#define NP   (Bq * Tq)        // 65536 positions
#define OUTW 416
#define EPSL 1e-5f

typedef __attribute__((ext_vector_type(2))) float v2f;
typedef __attribute__((ext_vector_type(8))) float v8f;

// ---------------------------------------------------------------------------
// Kernel 1: per-row segment analysis -> blend weight w and neighbor id nid
// One 1024-thread workgroup per batch row; row lives in LDS.
// ---------------------------------------------------------------------------
__global__ __launch_bounds__(1024)
void seg_blur_kernel(const int* __restrict__ ids_g,
                     float* __restrict__ wbuf, int* __restrict__ nidbuf) {
    __shared__ int s_ids[Tq];
    __shared__ int s_segstart[Tq];
    __shared__ int s_scan[1024];
    __shared__ int s_nseg;

    const int b   = blockIdx.x;
    const int tid = threadIdx.x;
    const int* row = ids_g + (size_t)b * Tq;

    for (int i = tid; i < Tq; i += 1024) s_ids[i] = row[i];
    __syncthreads();

    // run-start flags for this thread's 4 positions
    const int t0 = tid * 4;
    int fl[4]; int c = 0;
    #pragma unroll
    for (int i = 0; i < 4; ++i) {
        int t = t0 + i;
        fl[i] = (t == 0) || (s_ids[t] != s_ids[t - 1]);
        c += fl[i];
    }

    // inclusive prefix sum over 1024 per-thread counts (Hillis-Steele)
    s_scan[tid] = c;
    __syncthreads();
    for (int off = 1; off < 1024; off <<= 1) {
        int v = (tid >= off) ? s_scan[tid - off] : 0;
        __syncthreads();
        s_scan[tid] += v;
        __syncthreads();
    }
    const int incl = s_scan[tid];
    int base = incl - c;

    int segidx[4];
    #pragma unroll
    for (int i = 0; i < 4; ++i) {
        if (fl[i]) { s_segstart[base] = t0 + i; base++; }
        segidx[i] = base - 1;
    }
    if (tid == 1023) s_nseg = incl;
    __syncthreads();
    const int nseg = s_nseg;

    #pragma unroll
    for (int i = 0; i < 4; ++i) {
        const int t = t0 + i;
        const int s = segidx[i];
        const int seg_start = s_segstart[s];
        const int seg_end   = (s + 1 < nseg) ? s_segstart[s + 1] : Tq;
        const int dur = seg_end - seg_start;

        int prev_dur, prev_id;
        if (s > 0) { prev_dur = seg_start - s_segstart[s - 1]; prev_id = s_ids[seg_start - 1]; }
        else       { prev_dur = 0;                             prev_id = s_ids[0]; }

        int next_dur, next_id;
        if (s + 1 < nseg) {
            int ne = (s + 2 < nseg) ? s_segstart[s + 2] : Tq;
            next_dur = ne - seg_end;
            next_id  = s_ids[seg_end];
        } else { next_dur = 0; next_id = s_ids[Tq - 1]; }

        // LEFT boundary blend (into head of this segment)
        float radL = 0.3f * (float)min(prev_dur, dur);
        bool validL = (seg_start > 0) && (radL >= 0.5f);
        int  riL = max(1, (int)rintf(radL));
        int  reL = min(Tq, seg_start + riL);
        bool inL = (t >= seg_start) && (t < reL) && validL;
        float wL = inL ? fminf(0.5f, 0.5f * (float)(reL - t) / (float)riL) : 0.0f;

        // RIGHT boundary blend (into tail of this segment)
        float radR = 0.3f * (float)min(dur, next_dur);
        bool validR = (seg_end < Tq) && (radR >= 0.5f);
        int  riR = max(1, (int)rintf(radR));
        int  lsR = max(0, seg_end - riR);
        bool inR = (t >= lsR) && (t < seg_end) && validR;
        float wR = inR ? fminf(0.5f, 0.5f * (float)(t - lsR + 1) / (float)riR) : 0.0f;

        float w = fmaxf(wL, wR);
        int nid = (wR > wL) ? next_id : prev_id;
        if (!(w > 0.0f)) nid = s_ids[t];

        wbuf  [(size_t)b * Tq + t] = w;
        nidbuf[(size_t)b * Tq + t] = nid;
    }
}

// ---------------------------------------------------------------------------
// Kernel 2: 128-wide layernorms of x_t and prior_mel. One wave per position.
// ---------------------------------------------------------------------------
__global__ __launch_bounds__(256)
void ln128_kernel(const float* __restrict__ xt, const float* __restrict__ pr,
                  const float* __restrict__ g0, const float* __restrict__ b0,
                  const float* __restrict__ g1, const float* __restrict__ b1,
                  float* __restrict__ out) {
    const int wave = (blockIdx.x * blockDim.x + threadIdx.x) >> 5;  // position
    const int lane = threadIdx.x & 31;
    const size_t ibase = (size_t)wave * 128 + lane * 4;
    float* orow = out + (size_t)wave * OUTW + lane * 4;

    #pragma unroll
    for (int sel = 0; sel < 2; ++sel) {
        const float* src = sel ? pr : xt;
        const float* g   = sel ? g1 : g0;
        const float* bb  = sel ? b1 : b0;
        float4 a = *(const float4*)(src + ibase);
        float s = a.x + a.y + a.z + a.w;
        #pragma unroll
        for (int off = 16; off >= 1; off >>= 1) s += __shfl_xor(s, off, 32);
        float mean = s * (1.0f / 128.0f);
        float d0 = a.x - mean, d1 = a.y - mean, d2 = a.z - mean, d3 = a.w - mean;
        float vs = d0 * d0 + d1 * d1 + d2 * d2 + d3 * d3;
        #pragma unroll
        for (int off = 16; off >= 1; off >>= 1) vs += __shfl_xor(vs, off, 32);
        float inv = rsqrtf(vs * (1.0f / 128.0f) + EPSL);
        float4 gv = *(const float4*)(g + lane * 4);
        float4 bv = *(const float4*)(bb + lane * 4);
        float4 o;
        o.x = d0 * inv * gv.x + bv.x;
        o.y = d1 * inv * gv.y + bv.y;
        o.z = d2 * inv * gv.z + bv.z;
        o.w = d3 * inv * gv.w + bv.w;
        *(float4*)(orow + sel * 128) = o;
    }
}

// ---------------------------------------------------------------------------
// Kernel 3: f0 MLP via WMMA f32 16x16x4.
// One wave handles 16 positions: A = silu(f0*w1+b1) built directly in the
// 16x4-f32 A-fragment layout; B = w2 (LDS); D = 16x64 accumulated over 16
// K-chunks x 4 N-tiles (64 v_wmma per wave), then 64-wide LN in-registers.
// Sigmoid uses hardware V_RCP_F32 (TRANS pipe, co-executes with WMMA)
// instead of the 10-op IEEE division expansion.
// ---------------------------------------------------------------------------
__global__ __launch_bounds__(256)
void f0_wmma_kernel(const float* __restrict__ f0,
                    const float* __restrict__ w1, const float* __restrict__ b1,
                    const float* __restrict__ w2, const float* __restrict__ b2,
                    const float* __restrict__ g,  const float* __restrict__ bb,
                    float* __restrict__ out) {
    __shared__ float s_w2[64 * 64];
    for (int i = threadIdx.x; i < 64 * 64; i += blockDim.x) s_w2[i] = w2[i];
    __syncthreads();

    const int lane  = threadIdx.x & 31;
    const int half  = lane >> 4;       // 0: K%4 in {0,1};  1: K%4 in {2,3}
    const int l16   = lane & 15;
    const int wave  = (blockIdx.x * blockDim.x + threadIdx.x) >> 5;
    const int pbase = wave * 16;

    // A fragments: h[position = pbase+l16][K = 4k + 2*half + j]
    const float fp = f0[pbase + l16];
    float hA[32];
    #pragma unroll
    for (int k = 0; k < 16; ++k) {
        #pragma unroll
        for (int j = 0; j < 2; ++j) {
            int kk = 4 * k + 2 * half + j;
            float z = fp * w1[kk] + b1[kk];
            // silu(z) = z * sigmoid(z); v_rcp_f32 (~1 ulp) instead of IEEE div
            hA[2 * k + j] = z * __builtin_amdgcn_rcpf(1.0f + __expf(-z));
        }
    }

    v8f acc[4];
    #pragma unroll
    for (int nt = 0; nt < 4; ++nt) {
        v8f c = (v8f){0.f, 0.f, 0.f, 0.f, 0.f, 0.f, 0.f, 0.f};
        #pragma unroll
        for (int k = 0; k < 16; ++k) {
            v2f A = {hA[2 * k], hA[2 * k + 1]};
            const int kk0 = 4 * k + 2 * half;
            v2f Bf = {s_w2[kk0 * 64 + nt * 16 + l16],
                      s_w2[(kk0 + 1) * 64 + nt * 16 + l16]};
            c = __builtin_amdgcn_wmma_f32_16x16x4_f32(
                    false, A, false, Bf, (short)0, c, false, false);
        }
        acc[nt] = c;
    }

    float b2v[4], gv[4], bbv[4];
    #pragma unroll
    for (int nt = 0; nt < 4; ++nt) {
        int n = nt * 16 + l16;
        b2v[nt] = b2[n]; gv[nt] = g[n]; bbv[nt] = bb[n];
    }

    // D layout: lane -> N=l16, vgpr r -> M = 8*half + r. 64-wide LN per row
    // reduces across the 16 lanes of each half (xor offsets 8..1).
    #pragma unroll
    for (int r = 0; r < 8; ++r) {
        float e[4]; float s = 0.f;
        #pragma unroll
        for (int nt = 0; nt < 4; ++nt) { e[nt] = acc[nt][r] + b2v[nt]; s += e[nt]; }
        #pragma unroll
        for (int off = 8; off >= 1; off >>= 1) s += __shfl_xor(s, off, 32);
        float mean = s * (1.0f / 64.0f);
        float vs = 0.f;
        #pragma unroll
        for (int nt = 0; nt < 4; ++nt) { float d = e[nt] - mean; vs += d * d; }
        #pragma unroll
        for (int off = 8; off >= 1; off >>= 1) vs += __shfl_xor(vs, off, 32);
        float inv = rsqrtf(vs * (1.0f / 64.0f) + EPSL);
        const int p = pbase + half * 8 + r;
        float* op = out + (size_t)p * OUTW + 256;
        #pragma unroll
        for (int nt = 0; nt < 4; ++nt) {
            int n = nt * 16 + l16;
            op[n] = (e[nt] - mean) * inv * gv[nt] + bbv[nt];
        }
    }
}

// ---------------------------------------------------------------------------
// Kernel 4: blurred phoneme embed + LN(64), voicing embed + LN(32).
// One wave per position.
// ---------------------------------------------------------------------------
__global__ __launch_bounds__(256)
void phv_kernel(const int* __restrict__ ids, const float* __restrict__ wbuf,
                const int* __restrict__ nidbuf, const float* __restrict__ table,
                const float* __restrict__ g_ph, const float* __restrict__ b_ph,
                const int* __restrict__ voicing, const float* __restrict__ vtab,
                const float* __restrict__ g_v, const float* __restrict__ b_v,
                float* __restrict__ out) {
    const int p    = (blockIdx.x * blockDim.x + threadIdx.x) >> 5;
    const int lane = threadIdx.x & 31;
    float* orow = out + (size_t)p * OUTW;

    // ---- phoneme: 64 dims, 2 per lane ----
    const int   id  = ids[p];
    const int   nid = nidbuf[p];
    const float w   = wbuf[p];
    float2 e  = *(const float2*)(table + (size_t)id  * 64 + lane * 2);
    float2 ne = *(const float2*)(table + (size_t)nid * 64 + lane * 2);
    float m0 = (1.0f - w) * e.x + w * ne.x;
    float m1 = (1.0f - w) * e.y + w * ne.y;
    float s = m0 + m1;
    #pragma unroll
    for (int off = 16; off >= 1; off >>= 1) s += __shfl_xor(s, off, 32);
    float mean = s * (1.0f / 64.0f);
    float d0 = m0 - mean, d1 = m1 - mean;
    float vs = d0 * d0 + d1 * d1;
    #pragma unroll
    for (int off = 16; off >= 1; off >>= 1) vs += __shfl_xor(vs, off, 32);
    float inv = rsqrtf(vs * (1.0f / 64.0f) + EPSL);
    orow[320 + lane * 2 + 0] = d0 * inv * g_ph[lane * 2 + 0] + b_ph[lane * 2 + 0];
    orow[320 + lane * 2 + 1] = d1 * inv * g_ph[lane * 2 + 1] + b_ph[lane * 2 + 1];

    // ---- voicing: 32 dims, 1 per lane ----
    const int vv = voicing[p];
    float ve = vtab[vv * 32 + lane];
    float sv = ve;
    #pragma unroll
    for (int off = 16; off >= 1; off >>= 1) sv += __shfl_xor(sv, off, 32);
    float vmean = sv * (1.0f / 32.0f);
    float vd = ve - vmean;
    float vvs = vd * vd;
    #pragma unroll
    for (int off = 16; off >= 1; off >>= 1) vvs += __shfl_xor(vvs, off, 32);
    float vinv = rsqrtf(vvs * (1.0f / 32.0f) + EPSL);
    orow[384 + lane] = vd * vinv * g_v[lane] + b_v[lane];
}

// ---------------------------------------------------------------------------
extern "C" void kernel_launch(void* const* d_in, const int* in_sizes, int n_in,
                              void* d_out, int out_size, void* d_ws, size_t ws_size,
                              hipStream_t stream) {
    const float* x_t     = (const float*)d_in[0];
    const float* prior   = (const float*)d_in[1];
    const float* f0      = (const float*)d_in[2];
    const int*   voicing = (const int*)  d_in[3];
    const int*   ph_ids  = (const int*)  d_in[4];
    const float* table   = (const float*)d_in[5];
    const float* f0_w1   = (const float*)d_in[6];
    const float* f0_b1   = (const float*)d_in[7];
    const float* f0_w2   = (const float*)d_in[8];
    const float* f0_b2   = (const float*)d_in[9];
    const float* vtab    = (const float*)d_in[10];
    const float* g_xt    = (const float*)d_in[11];
    const float* b_xt    = (const float*)d_in[12];
    const float* g_pr    = (const float*)d_in[13];
    const float* b_pr    = (const float*)d_in[14];
    const float* g_f0    = (const float*)d_in[15];
    const float* b_f0    = (const float*)d_in[16];
    const float* g_ph    = (const float*)d_in[17];
    const float* b_ph    = (const float*)d_in[18];
    const float* g_v     = (const float*)d_in[19];
    const float* b_v     = (const float*)d_in[20];

    float* out   = (float*)d_out;
    float* wbuf  = (float*)d_ws;
    int*   nidbf = (int*)((char*)d_ws + (size_t)NP * sizeof(float));

    // 1) segment analysis + blur weights (one WG per batch row)
    seg_blur_kernel<<<Bq, 1024, 0, stream>>>(ph_ids, wbuf, nidbf);

    // 2) 128-wide layernorms (x_t, prior): 1 wave/pos, 8 waves/block
    ln128_kernel<<<NP / 8, 256, 0, stream>>>(x_t, prior, g_xt, b_xt, g_pr, b_pr, out);

    // 3) f0 MLP via WMMA: 16 pos/wave, 8 waves/block -> 128 pos/block
    f0_wmma_kernel<<<NP / 128, 256, 0, stream>>>(f0, f0_w1, f0_b1, f0_w2, f0_b2,
                                                 g_f0, b_f0, out);

    // 4) phoneme blend + voicing: 1 wave/pos
    phv_kernel<<<NP / 8, 256, 0, stream>>>(ph_ids, wbuf, nidbf, table, g_ph, b_ph,
                                           voicing, vtab, g_v, b_v, out);
}